// GridErasing_17875653886410
// MI455X (gfx1250) — compile-verified
//
#include <hip/hip_runtime.h>

// GridErasing: out = mask(h,w) ? noise : x   over (C,H,W) fp32
// Zero-FLOP streaming op -> bandwidth-roofline kernel for MI455X (23.3 TB/s).
//  * float4 (global_load/store_b128) non-temporal accesses (streams, no L2 reuse)
//  * mask computed once per (h,w) thread, reused across all 256 channels
//  * wave32 __ballot: all-copy waves skip the noise stream entirely
//  * exec-masked (per-lane predicated) noise load in mixed waves: unmasked
//    lane groups contribute no addresses -> their noise cachelines are skipped

typedef __attribute__((ext_vector_type(4))) float v4f;

#define C_  256
#define H_  1024
#define W_  1024
#define WQ_ (W_ / 4)          // float4 columns per row == 256

__device__ __forceinline__ int iclamp(int v, int lo, int hi) {
    return v < lo ? lo : (v > hi ? hi : v);
}

__global__ __launch_bounds__(256) void grid_erase_kernel(
    const float* __restrict__ x,
    const float* __restrict__ noise,
    const unsigned char* __restrict__ cell,   // jax bool array: 1 byte/elem
    const int* __restrict__ p_stride,
    const int* __restrict__ p_grid,
    const int* __restrict__ p_oh,
    const int* __restrict__ p_ow,
    float* __restrict__ out)
{
    const int stride = p_stride[0];
    const int gsz    = p_grid[0];
    const int oh     = p_oh[0];
    const int ow     = p_ow[0];
    // NCH = len(range(oh, H, stride)) = ceil((H-oh)/stride); same for NCW
    const int nch = (H_ - oh + stride - 1) / stride;
    const int ncw = (W_ - ow + stride - 1) / stride;

    const int gid = blockIdx.x * blockDim.x + threadIdx.x;  // 0 .. H_*WQ_-1
    const int h   = gid >> 8;                // gid / WQ_ (WQ_ == 256)
    const int wq  = gid & (WQ_ - 1);
    const int w0  = wq << 2;

    // Row predicate (shared by this thread's 4 columns).
    // Matches reference: in_h = (ri>=0) & (ri%stride < gsz); ci = clip(ri//stride,0,nch-1)
    // (for ri<0 the predicate is false; trunc vs floor division both clamp to 0)
    const int  ri   = h - oh;
    const int  qi   = ri / stride;
    const int  remi = ri - qi * stride;
    const bool in_h = (ri >= 0) && (remi < gsz);
    const int  ci   = iclamp(qi, 0, nch - 1);

    bool m[4];
#pragma unroll
    for (int j = 0; j < 4; ++j) {
        const int  rj   = (w0 + j) - ow;
        const int  qj   = rj / stride;
        const int  remj = rj - qj * stride;
        const bool in_w = (rj >= 0) && (remj < gsz);
        const int  cj   = iclamp(qj, 0, ncw - 1);
        m[j] = in_h && in_w && (cell[ci * ncw + cj] != 0);
    }
    const bool m0 = m[0], m1 = m[1], m2 = m[2], m3 = m[3];
    const bool need_noise = m0 | m1 | m2 | m3;

    const v4f* __restrict__ x4 = (const v4f*)x;
    const v4f* __restrict__ n4 = (const v4f*)noise;
    v4f*       __restrict__ o4 = (v4f*)out;

    const int plane4 = H_ * WQ_;             // float4 elements per channel
    int idx = h * WQ_ + wq;

    // Wave-uniform split (wave32 ballot): waves with no erased lane run the
    // pure streaming copy and never touch the noise stream.
    if (__ballot(need_noise) != 0ull) {
#pragma unroll 4
        for (int c = 0; c < C_; ++c, idx += plane4) {
            v4f xv = __builtin_nontemporal_load(x4 + idx);
            v4f nv = xv;
            // Exec-masked load: only lanes with an erased element fetch noise,
            // so untouched cachelines within a mixed wave are skipped.
            if (need_noise) {
                nv = __builtin_nontemporal_load(n4 + idx);
            }
            v4f ov;
            ov.x = m0 ? nv.x : xv.x;
            ov.y = m1 ? nv.y : xv.y;
            ov.z = m2 ? nv.z : xv.z;
            ov.w = m3 ? nv.w : xv.w;
            __builtin_nontemporal_store(ov, o4 + idx);
        }
    } else {
#pragma unroll 8
        for (int c = 0; c < C_; ++c, idx += plane4) {
            v4f xv = __builtin_nontemporal_load(x4 + idx);
            __builtin_nontemporal_store(xv, o4 + idx);
        }
    }
}

extern "C" void kernel_launch(void* const* d_in, const int* in_sizes, int n_in,
                              void* d_out, int out_size, void* d_ws, size_t ws_size,
                              hipStream_t stream) {
    const float*         x      = (const float*)d_in[0];
    const float*         noise  = (const float*)d_in[1];
    const unsigned char* cell   = (const unsigned char*)d_in[2];
    const int*           p_str  = (const int*)d_in[3];
    const int*           p_gsz  = (const int*)d_in[4];
    const int*           p_oh   = (const int*)d_in[5];
    const int*           p_ow   = (const int*)d_in[6];
    float*               out    = (float*)d_out;

    const int n_threads = H_ * WQ_;          // one thread per (h, float4-col)
    const int block = 256;                   // 8 wave32s
    const int grid  = n_threads / block;     // 1024 blocks, each loops over C

    grid_erase_kernel<<<grid, block, 0, stream>>>(
        x, noise, cell, p_str, p_gsz, p_oh, p_ow, out);
}